// GraphGenerator_61598420959302
// MI455X (gfx1250) — compile-verified
//
#include <hip/hip_runtime.h>
#include <hip/hip_bf16.h>

typedef __attribute__((ext_vector_type(2))) float v2f;
typedef __attribute__((ext_vector_type(8))) float v8f;

// lane L <-> lane L^16 exchange: ds_swizzle group-of-32, xor=0x10, and=0x1f
static __device__ __forceinline__ float swap16(float v) {
    return __builtin_bit_cast(float,
        __builtin_amdgcn_ds_swizzle(__builtin_bit_cast(int, v), 0x401f));
}
// single-instruction ReLU (inputs are finite): median(x, 0, FLT_MAX)
static __device__ __forceinline__ float relu1(float v) {
    return __builtin_amdgcn_fmed3f(v, 0.0f, 3.402823466e+38f);
}
// force a uniform-address load to stay scalar (SGPR) — stops the compiler from
// synthesizing per-lane addressed vector loads for weight tables
static __device__ __forceinline__ float sloadf(const float* p) {
    float v = *p;
    return __builtin_bit_cast(float, __builtin_amdgcn_readfirstlane(__builtin_bit_cast(int, v)));
}

// One wave (32 lanes) handles 32 edges = two 16-edge WMMA tiles.
//
// A (16x4 f32): row-permuted W1 with fused bias column.
//   A-row m: m in [0,5) -> W1 row m ; m in [8,13) -> W1 row m-3 ; else 0
//   A cols:  K0..K2 = W1[row][0..2], K3 = b1[row]
// B (4x16 f32): per edge column n = lane&15:
//   rows K0 = x_src, K1 = x_tgt, K2 = edge_attr, K3 = 1.0 (bias multiplier)
// D: lane's D[r] = pre-activation of hidden row (r + 8*half) for edge n;
//    live rows are slots r = 0..4 in both halves (lo: h0-h4, hi: h5-h9).
//
// Output needs only o1 - o0 (softmax of 2 == sigmoid of the logit diff); all
// post-ReLU math is linear => single dot with wd = W2[0]-W2[1], directions
// summed first, one cross-half swizzle per tile, one exp per lane.
__global__ __launch_bounds__(256) void edge_mlp_wmma_f32(
    const float* __restrict__ x,
    const float* __restrict__ edge_attr,
    const float* __restrict__ edge_attr_T,
    const int*   __restrict__ edge_index,   // [2, E] row-major
    const float* __restrict__ W1,           // [10,3]
    const float* __restrict__ b1,           // [10]
    const float* __restrict__ W2,           // [2,10]
    const float* __restrict__ b2,           // [2]
    float* __restrict__ out,                // [E]
    int E, int nWaves)
{
    const int wave = blockIdx.x * (blockDim.x >> 5) + (threadIdx.x >> 5);
    if (wave >= nWaves) return;               // wave-uniform guard, EXEC stays all-ones

    const int lane = threadIdx.x & 31;
    const int n    = lane & 15;
    const int half = lane >> 4;

    // ---- phase 1: all weight-table scalars, constant-index, pinned uniform ----
    float w2lo0[5], w2lo1[5], w2hi0[5], w2hi1[5];
#pragma unroll
    for (int r = 0; r < 5; ++r) {
        w2lo0[r] = sloadf(W2 + r);            // W2[0][r]
        w2lo1[r] = sloadf(W2 + 10 + r);       // W2[1][r]
        w2hi0[r] = sloadf(W2 + 5 + r);        // W2[0][r+5]
        w2hi1[r] = sloadf(W2 + 15 + r);       // W2[1][r+5]
    }
    const float b20 = sloadf(b2 + 0);
    const float b21 = sloadf(b2 + 1);

    // ---- phase 2: uniform arithmetic ----
    float wdlo[5], wdhi[5];
#pragma unroll
    for (int r = 0; r < 5; ++r) {
        wdlo[r] = w2lo0[r] - w2lo1[r];
        wdhi[r] = w2hi0[r] - w2hi1[r];
    }
    const float bdiff = b20 - b21;

    const int e0  = wave * 32 + n;
    const int e1  = e0 + 16;
    const int ec0 = (e0 < E) ? e0 : (E - 1);
    const int ec1 = (e1 < E) ? e1 : (E - 1);

    // ---- edge indices ----
    const int src0 = edge_index[ec0];
    const int tgt0 = edge_index[E + ec0];
    const int src1 = edge_index[ec1];
    const int tgt1 = edge_index[E + ec1];

    // ---- address-selected loads: lo lanes gather x (L2-resident), hi lanes stream attr ----
    const float* p10 = half ? (edge_attr   + ec0) : (x + src0);
    const float* p20 = half ? (edge_attr_T + ec0) : (x + tgt0);
    const float* p11 = half ? (edge_attr   + ec1) : (x + src1);
    const float* p21 = half ? (edge_attr_T + ec1) : (x + tgt1);
    const float g10 = *p10, g20 = *p20, g11 = *p11, g21 = *p21;

    // ---- A operand: row-permuted W1 + fused bias column (lane-indexed, cached) ----
    const int  m     = n;
    const bool loOK  = (m < 5);
    const bool hiOK  = (m >= 8) && (m < 13);
    const int  row   = loOK ? m : (hiOK ? (m - 3) : 0);   // clamped, no OOB
    const bool rowOK = loOK || hiOK;
    const float w1_0 = W1[row * 3 + 0];
    const float w1_1 = W1[row * 3 + 1];
    const float w1_2 = W1[row * 3 + 2];
    const float b1r  = b1[row];
    v2f A;
    A.x = rowOK ? (half ? w1_2 : w1_0) : 0.0f;   // K = 2*half
    A.y = rowOK ? (half ? b1r  : w1_1) : 0.0f;   // K = 1+2*half (K3 column = bias)

    // ---- B operands: K3 row = 1.0 multiplies the bias column ----
    v2f Bd0, Bt0, Bd1, Bt1;
    Bd0.x = g10;  Bd0.y = half ? 1.0f : g20;
    Bt0.x = g20;  Bt0.y = half ? 1.0f : g10;
    Bd1.x = g11;  Bd1.y = half ? 1.0f : g21;
    Bt1.x = g21;  Bt1.y = half ? 1.0f : g11;

    // ---- layer 1 + bias on the matrix pipe (exact f32) ----
    v8f cz = {};
    v8f h1a = __builtin_amdgcn_wmma_f32_16x16x4_f32(false, A, false, Bd0, (short)0, cz, false, false);
    v8f h1b = __builtin_amdgcn_wmma_f32_16x16x4_f32(false, A, false, Bt0, (short)0, cz, false, false);
    v8f h2a = __builtin_amdgcn_wmma_f32_16x16x4_f32(false, A, false, Bd1, (short)0, cz, false, false);
    v8f h2b = __builtin_amdgcn_wmma_f32_16x16x4_f32(false, A, false, Bt1, (short)0, cz, false, false);

    // ---- ReLU + differenced layer 2: t = sum_m wd[m] * (relu_d1 + relu_d2) ----
    float t0 = 0.f, t1 = 0.f;
#pragma unroll
    for (int r = 0; r < 5; ++r) {
        const float wd = half ? wdhi[r] : wdlo[r];   // phase 3: one select per slot
        t0 = fmaf(wd, relu1(h1a[r]) + relu1(h1b[r]), t0);
        t1 = fmaf(wd, relu1(h2a[r]) + relu1(h2b[r]), t1);
    }

    // ---- cross-half reduction (rows 0-4 vs 5-9): one swizzle per tile ----
    t0 += swap16(t0);
    t1 += swap16(t1);

    // o0 - o1 = 0.5 * t_full + (b2[0]-b2[1]); each lane finishes only its own edge
    const float tsel = half ? t1 : t0;      // lo lanes own tile0, hi lanes tile1
    const float d    = fmaf(0.5f, tsel, bdiff);

    // softmax(axis=1)[:,1] == sigmoid(o1-o0) = 1/(1+exp(o0-o1))
    const float res = __builtin_amdgcn_rcpf(1.0f + __expf(d));

    // fully-coalesced store: lane L -> edge wave*32 + L
    const int eo = wave * 32 + lane;
    if (eo < E) out[eo] = res;
}

extern "C" void kernel_launch(void* const* d_in, const int* in_sizes, int n_in,
                              void* d_out, int out_size, void* d_ws, size_t ws_size,
                              hipStream_t stream) {
    (void)n_in; (void)d_ws; (void)ws_size; (void)out_size;
    const float* x    = (const float*)d_in[0];
    // d_in[1] = token, unused (token_enabled=False)
    const float* ea   = (const float*)d_in[2];
    const float* eaT  = (const float*)d_in[3];
    const int*   eidx = (const int*)d_in[4];
    const float* W1   = (const float*)d_in[5];
    const float* b1   = (const float*)d_in[6];
    const float* W2   = (const float*)d_in[7];
    const float* b2   = (const float*)d_in[8];
    float* out = (float*)d_out;

    const int E      = in_sizes[2];            // N_EDGES
    const int nWaves = (E + 31) / 32;          // 32 edges per wave
    const int wavesPerBlock = 8;               // 256 threads = 8 wave32
    const int blocks = (nWaves + wavesPerBlock - 1) / wavesPerBlock;

    edge_mlp_wmma_f32<<<blocks, 256, 0, stream>>>(x, ea, eaT, eidx, W1, b1, W2, b2,
                                                  out, E, nWaves);
}